// SALayerSingleHead_2070174237152
// MI455X (gfx1250) — compile-verified
//
#include <hip/hip_runtime.h>
#include <math.h>
#include <stdint.h>

// SA_Layer (single-head self-attention + FFN), B=8 N=2048 C=128, fp32 via
// V_WMMA_F32_16X16X4_F32.  Attention matrix is never materialized: pass 1
// computes per-row softmax stats with WMMA energy tiles + online merge,
// pass 2 recomputes energy tiles and feeds P tiles straight into the
// x_v * attention WMMA accumulation, with V tiles double-buffered in LDS
// via GLOBAL_LOAD_ASYNC_TO_LDS_B128 (ASYNCcnt).  Plain GEMM kernels use
// 2x2 register blocking so each WMMA costs ~1 fragment load instead of 2.

#define BATCH 8
#define NPTS  2048
#define CCH   128
#define DQK   32
#define FDIM  512
#define BN_EPSF 1e-5f
#define SQRT_INV_C 0.08838834764831845f  // 1/sqrt(128)

typedef __attribute__((ext_vector_type(2))) float v2f;
typedef __attribute__((ext_vector_type(8))) float v8f;

__device__ __forceinline__ v8f wmma4(v2f a, v2f b, v8f c) {
  // D = A(16x4 f32) * B(4x16 f32) + C(16x16 f32)
  return __builtin_amdgcn_wmma_f32_16x16x4_f32(
      /*neg_a=*/false, a, /*neg_b=*/false, b,
      /*c_mod=*/(short)0, c, /*reuse_a=*/false, /*reuse_b=*/false);
}

// Fragment load: row-major matrix, lane holds M[lo][k+h2 .. k+h2+1] (8B aligned)
__device__ __forceinline__ v2f ldrow(const float* __restrict__ p, int ld,
                                     int lo, int h2, int k) {
  return *(const v2f*)(p + (size_t)lo * ld + k + h2);
}

// ---------------------------------------------------------------------------
// Kernel 1: QKV projections, 2x2-blocked.  out[n,o] = sum_c x[n,c]*W[o,c]
// grid (N/32, 6, B), 32 threads (1 wave).  Pair 0->Wq, 1->Wk, 2..5->Wv.
// ---------------------------------------------------------------------------
__global__ __launch_bounds__(32) void qkv_kernel(
    const float* __restrict__ x, const float* __restrict__ Wq,
    const float* __restrict__ Wk, const float* __restrict__ Wv,
    const float* __restrict__ bv, float* __restrict__ Q,
    float* __restrict__ Kf, float* __restrict__ V) {
  const int lane = threadIdx.x & 31;
  const int lo = lane & 15, half = lane >> 4, h2 = half * 2;
  const int np = blockIdx.x, p = blockIdx.y, b = blockIdx.z;
  const int n0 = np * 32;

  const float* W; float* out; int ldo, orow0; bool addb;
  if (p == 0)      { W = Wq; out = Q  + (size_t)b*NPTS*DQK; ldo = DQK; orow0 = 0;        addb = false; }
  else if (p == 1) { W = Wk; out = Kf + (size_t)b*NPTS*DQK; ldo = DQK; orow0 = 0;        addb = false; }
  else             { W = Wv; out = V  + (size_t)b*NPTS*CCH; ldo = CCH; orow0 = (p-2)*32; addb = true;  }

  const float* A0 = x + ((size_t)b*NPTS + n0) * CCH;
  const float* A1 = A0 + 16 * CCH;
  const float* B0 = W + (size_t)orow0 * CCH;
  const float* B1 = B0 + 16 * CCH;

  v8f acc[2][2] = {};
#pragma unroll 4
  for (int k = 0; k < CCH; k += 4) {
    v2f a0 = ldrow(A0, CCH, lo, h2, k), a1 = ldrow(A1, CCH, lo, h2, k);
    v2f b0 = ldrow(B0, CCH, lo, h2, k), b1 = ldrow(B1, CCH, lo, h2, k);
    acc[0][0] = wmma4(a0, b0, acc[0][0]);
    acc[0][1] = wmma4(a0, b1, acc[0][1]);
    acc[1][0] = wmma4(a1, b0, acc[1][0]);
    acc[1][1] = wmma4(a1, b1, acc[1][1]);
  }

#pragma unroll
  for (int tj = 0; tj < 2; ++tj) {
    const int oc = orow0 + tj*16 + lo;
    const float bias = addb ? bv[oc] : 0.0f;
#pragma unroll
    for (int ti = 0; ti < 2; ++ti)
#pragma unroll
      for (int i = 0; i < 8; ++i) {
        int nr = n0 + ti*16 + half*8 + i;
        out[(size_t)nr * ldo + oc] = acc[ti][tj][i] + bias;
      }
  }
}

// ---------------------------------------------------------------------------
// Kernel 2: per-row softmax stats over m of E[n,m] = (Q[n,:].K[m,:])/sqrt(C).
// grid (N/16, B), 256 threads (8 waves); Q fragment hoisted out of m loop.
// ---------------------------------------------------------------------------
__global__ __launch_bounds__(256) void stats_kernel(
    const float* __restrict__ Q, const float* __restrict__ Kf,
    float* __restrict__ rowmax, float* __restrict__ rowsum) {
  const int lane = threadIdx.x & 31, wave = threadIdx.x >> 5;
  const int lo = lane & 15, half = lane >> 4, h2 = half * 2;
  const int nt = blockIdx.x, b = blockIdx.y;
  const float* Qrow = Q + ((size_t)b*NPTS + nt*16) * DQK;
  const float* Kb   = Kf + (size_t)b*NPTS*DQK;

  v2f qf[8];
#pragma unroll
  for (int kk = 0; kk < 8; ++kk) qf[kk] = ldrow(Qrow, DQK, lo, h2, kk*4);

  float mx[8], sm[8];
#pragma unroll
  for (int i = 0; i < 8; ++i) { mx[i] = -3.0e38f; sm[i] = 0.0f; }

  for (int mt = wave; mt < NPTS/16; mt += 8) {
    const float* Brow = Kb + (size_t)mt*16*DQK;
    v8f e = {};
#pragma unroll
    for (int kk = 0; kk < 8; ++kk)
      e = wmma4(qf[kk], ldrow(Brow, DQK, lo, h2, kk*4), e);
#pragma unroll
    for (int i = 0; i < 8; ++i) {
      float v = e[i] * SQRT_INV_C;
      float M = fmaxf(mx[i], v);
      sm[i] = sm[i] * __expf(mx[i] - M) + __expf(v - M);
      mx[i] = M;
    }
  }
  // merge across the 16 columns held by each lane half
#pragma unroll
  for (int d = 1; d < 16; d <<= 1) {
#pragma unroll
    for (int i = 0; i < 8; ++i) {
      float omx = __shfl_xor(mx[i], d, 32);
      float osm = __shfl_xor(sm[i], d, 32);
      float M = fmaxf(mx[i], omx);
      sm[i] = sm[i] * __expf(mx[i] - M) + osm * __expf(omx - M);
      mx[i] = M;
    }
  }
  __shared__ float2 red[8][16];
  if (lo == 0) {
#pragma unroll
    for (int i = 0; i < 8; ++i) red[wave][half*8 + i] = make_float2(mx[i], sm[i]);
  }
  __syncthreads();
  if (threadIdx.x < 16) {
    int r = threadIdx.x;
    float M = red[0][r].x, S = red[0][r].y;
#pragma unroll
    for (int w = 1; w < 8; ++w) {
      float2 pr = red[w][r];
      float M2 = fmaxf(M, pr.x);
      S = S * __expf(M - M2) + pr.y * __expf(pr.x - M2);
      M = M2;
    }
    rowmax[(size_t)b*NPTS + nt*16 + r] = M;
    rowsum[(size_t)b*NPTS + nt*16 + r] = S;
  }
}

// ---------------------------------------------------------------------------
// Kernel 3: x_r[c,m] = sum_n V[n,c] * softmax(E)[n,m]; fused U = x - x_r^T.
// grid (N/16 over m, B), 32 threads.  K-tile fragment hoisted; V tiles
// (contiguous 8KB) double-buffered in LDS via async-to-LDS B128; P^T
// fragments reused across all 8 channel tiles.
// ---------------------------------------------------------------------------
__global__ __launch_bounds__(32) void attnv_kernel(
    const float* __restrict__ Q, const float* __restrict__ Kf,
    const float* __restrict__ V, const float* __restrict__ x,
    const float* __restrict__ rowmax, const float* __restrict__ rowsum,
    float* __restrict__ U) {
  const int lane = threadIdx.x & 31;
  const int lo = lane & 15, half = lane >> 4, h2 = half * 2;
  const int mt = blockIdx.x, b = blockIdx.y;
  const float* Qb   = Q + (size_t)b*NPTS*DQK;
  const float* Krow = Kf + ((size_t)b*NPTS + mt*16) * DQK;
  const float* Vb   = V + (size_t)b*NPTS*CCH;
  const float* rmx  = rowmax + (size_t)b*NPTS;
  const float* rsm  = rowsum + (size_t)b*NPTS;

  __shared__ __align__(16) float vtile[2][16 * CCH];  // 2 x 8KB V tiles
  __shared__ float pT[16 * 16];                       // [n_local][m_local]

  // K-tile B-operand fragments: invariant over the whole n sweep
  v2f kfr[8];
#pragma unroll
  for (int kk = 0; kk < 8; ++kk) kfr[kk] = ldrow(Krow, DQK, lo, h2, kk*4);

  v8f accC[8] = {};
  const uint32_t lane16 = (uint32_t)(lane * 16);

  // prefetch V tile 0 into buffer 0 (16 x B128 per wave = 8KB)
  {
    uint32_t lbase = (uint32_t)(uintptr_t)(&vtile[0][0]);
    uint64_t g = (uint64_t)(uintptr_t)Vb;
#pragma unroll
    for (int j = 0; j < 16; ++j)
      asm volatile("global_load_async_to_lds_b128 %0, %1, off"
                   :: "v"(lbase + (uint32_t)(j*512) + lane16),
                      "v"(g + (uint64_t)(j*512) + lane16)
                   : "memory");
  }

  for (int nt = 0; nt < NPTS/16; ++nt) {
    const int buf = nt & 1;
    if (nt + 1 < NPTS/16) {
      // issue next tile, then wait for the current tile's 16 issues
      uint32_t lbase = (uint32_t)(uintptr_t)(&vtile[buf ^ 1][0]);
      uint64_t g = (uint64_t)(uintptr_t)(Vb + (size_t)(nt + 1)*16*CCH);
#pragma unroll
      for (int j = 0; j < 16; ++j)
        asm volatile("global_load_async_to_lds_b128 %0, %1, off"
                     :: "v"(lbase + (uint32_t)(j*512) + lane16),
                        "v"(g + (uint64_t)(j*512) + lane16)
                     : "memory");
      asm volatile("s_wait_asynccnt 0x10" ::: "memory");
    } else {
      asm volatile("s_wait_asynccnt 0x0" ::: "memory");
    }

    // E tile: rows = n(ntile), cols = m(mtile)
    const float* Arow = Qb + (size_t)nt*16*DQK;
    v8f e = {};
#pragma unroll
    for (int kk = 0; kk < 8; ++kk)
      e = wmma4(ldrow(Arow, DQK, lo, h2, kk*4), kfr[kk], e);

#pragma unroll
    for (int i = 0; i < 8; ++i) {
      int nr = nt*16 + half*8 + i;
      pT[(half*8 + i) * 16 + lo] = __expf(e[i] * SQRT_INV_C - rmx[nr]) / rsm[nr];
    }
    __syncthreads();

    // x_r^T[m,c] += P^T[m,n] * V[n,c]; A fragment reused across 8 c-tiles
    const float* Vl = &vtile[buf][0];
#pragma unroll
    for (int kb = 0; kb < 16; kb += 4) {
      v2f a;
      a.x = pT[(kb + h2) * 16 + lo];
      a.y = pT[(kb + h2 + 1) * 16 + lo];
#pragma unroll
      for (int ct = 0; ct < 8; ++ct) {
        v2f bb;
        bb.x = Vl[(kb + h2) * CCH + ct*16 + lo];
        bb.y = Vl[(kb + h2 + 1) * CCH + ct*16 + lo];
        accC[ct] = wmma4(a, bb, accC[ct]);
      }
    }
    __syncthreads();
  }

  const float* xb = x + ((size_t)b*NPTS + mt*16) * CCH;
  float* Ub = U + ((size_t)b*NPTS + mt*16) * CCH;
#pragma unroll
  for (int ct = 0; ct < 8; ++ct)
#pragma unroll
    for (int i = 0; i < 8; ++i) {
      int m = half*8 + i, c = ct*16 + lo;
      Ub[(size_t)m*CCH + c] = xb[(size_t)m*CCH + c] - accC[ct][i];
    }
}

// ---------------------------------------------------------------------------
// Kernel 4: h = x + relu(BN1(Wt*U + bt)), 2x2-blocked.
// grid (N/32, C/32, B), 32 threads.
// ---------------------------------------------------------------------------
__global__ __launch_bounds__(32) void trans_kernel(
    const float* __restrict__ U, const float* __restrict__ Wt,
    const float* __restrict__ bt, const float* __restrict__ g1,
    const float* __restrict__ b1, const float* __restrict__ m1,
    const float* __restrict__ v1, const float* __restrict__ x,
    float* __restrict__ H) {
  const int lane = threadIdx.x & 31;
  const int lo = lane & 15, half = lane >> 4, h2 = half * 2;
  const int np = blockIdx.x, op = blockIdx.y, b = blockIdx.z;
  const int n0 = np * 32, o0 = op * 32;
  const float* A0 = U + ((size_t)b*NPTS + n0) * CCH;
  const float* A1 = A0 + 16 * CCH;
  const float* B0 = Wt + (size_t)o0 * CCH;
  const float* B1 = B0 + 16 * CCH;

  v8f acc[2][2] = {};
#pragma unroll 4
  for (int k = 0; k < CCH; k += 4) {
    v2f a0 = ldrow(A0, CCH, lo, h2, k), a1 = ldrow(A1, CCH, lo, h2, k);
    v2f b0 = ldrow(B0, CCH, lo, h2, k), b1 = ldrow(B1, CCH, lo, h2, k);
    acc[0][0] = wmma4(a0, b0, acc[0][0]);
    acc[0][1] = wmma4(a0, b1, acc[0][1]);
    acc[1][0] = wmma4(a1, b0, acc[1][0]);
    acc[1][1] = wmma4(a1, b1, acc[1][1]);
  }

  const float* xb = x + ((size_t)b*NPTS + n0) * CCH;
  float* Hb = H + ((size_t)b*NPTS + n0) * CCH;
#pragma unroll
  for (int tj = 0; tj < 2; ++tj) {
    const int o = o0 + tj*16 + lo;
    const float sc = g1[o] * rsqrtf(v1[o] + BN_EPSF);
    const float sh = b1[o] - m1[o] * sc;
    const float bo = bt[o];
#pragma unroll
    for (int ti = 0; ti < 2; ++ti)
#pragma unroll
      for (int i = 0; i < 8; ++i) {
        int n = ti*16 + half*8 + i;
        float t = (acc[ti][tj][i] + bo) * sc + sh;
        Hb[(size_t)n*CCH + o] = xb[(size_t)n*CCH + o] + fmaxf(t, 0.0f);
      }
  }
}

// ---------------------------------------------------------------------------
// Kernel 5: f = relu(W1*h + bf1), 2x2-blocked.  grid (N/32, 512/32, B).
// ---------------------------------------------------------------------------
__global__ __launch_bounds__(32) void ffn1_kernel(
    const float* __restrict__ H, const float* __restrict__ W1,
    const float* __restrict__ bf1, float* __restrict__ Ff) {
  const int lane = threadIdx.x & 31;
  const int lo = lane & 15, half = lane >> 4, h2 = half * 2;
  const int np = blockIdx.x, jp = blockIdx.y, b = blockIdx.z;
  const int n0 = np * 32, j0 = jp * 32;
  const float* A0 = H + ((size_t)b*NPTS + n0) * CCH;
  const float* A1 = A0 + 16 * CCH;
  const float* B0 = W1 + (size_t)j0 * CCH;
  const float* B1 = B0 + 16 * CCH;

  v8f acc[2][2] = {};
#pragma unroll 4
  for (int k = 0; k < CCH; k += 4) {
    v2f a0 = ldrow(A0, CCH, lo, h2, k), a1 = ldrow(A1, CCH, lo, h2, k);
    v2f b0 = ldrow(B0, CCH, lo, h2, k), b1 = ldrow(B1, CCH, lo, h2, k);
    acc[0][0] = wmma4(a0, b0, acc[0][0]);
    acc[0][1] = wmma4(a0, b1, acc[0][1]);
    acc[1][0] = wmma4(a1, b0, acc[1][0]);
    acc[1][1] = wmma4(a1, b1, acc[1][1]);
  }

  float* Fb = Ff + ((size_t)b*NPTS + n0) * FDIM;
#pragma unroll
  for (int tj = 0; tj < 2; ++tj) {
    const int j = j0 + tj*16 + lo;
    const float bj = bf1[j];
#pragma unroll
    for (int ti = 0; ti < 2; ++ti)
#pragma unroll
      for (int i = 0; i < 8; ++i) {
        int n = ti*16 + half*8 + i;
        Fb[(size_t)n*FDIM + j] = fmaxf(acc[ti][tj][i] + bj, 0.0f);
      }
  }
}

// ---------------------------------------------------------------------------
// Kernel 6: out = BN2(h + W2*f + bf2), 2x2-blocked.  grid (N/32, C/32, B).
// ---------------------------------------------------------------------------
__global__ __launch_bounds__(32) void ffn2_kernel(
    const float* __restrict__ Ff, const float* __restrict__ W2,
    const float* __restrict__ bf2, const float* __restrict__ H,
    const float* __restrict__ g2, const float* __restrict__ b2,
    const float* __restrict__ m2, const float* __restrict__ v2,
    float* __restrict__ out) {
  const int lane = threadIdx.x & 31;
  const int lo = lane & 15, half = lane >> 4, h2 = half * 2;
  const int np = blockIdx.x, op = blockIdx.y, b = blockIdx.z;
  const int n0 = np * 32, o0 = op * 32;
  const float* A0 = Ff + ((size_t)b*NPTS + n0) * FDIM;
  const float* A1 = A0 + 16 * FDIM;
  const float* B0 = W2 + (size_t)o0 * FDIM;
  const float* B1 = B0 + 16 * FDIM;

  v8f acc[2][2] = {};
#pragma unroll 4
  for (int k = 0; k < FDIM; k += 4) {
    v2f a0 = ldrow(A0, FDIM, lo, h2, k), a1 = ldrow(A1, FDIM, lo, h2, k);
    v2f b0 = ldrow(B0, FDIM, lo, h2, k), b1 = ldrow(B1, FDIM, lo, h2, k);
    acc[0][0] = wmma4(a0, b0, acc[0][0]);
    acc[0][1] = wmma4(a0, b1, acc[0][1]);
    acc[1][0] = wmma4(a1, b0, acc[1][0]);
    acc[1][1] = wmma4(a1, b1, acc[1][1]);
  }

  const float* Hb = H + ((size_t)b*NPTS + n0) * CCH;
  float* ob = out + ((size_t)b*NPTS + n0) * CCH;
#pragma unroll
  for (int tj = 0; tj < 2; ++tj) {
    const int o = o0 + tj*16 + lo;
    const float sc = g2[o] * rsqrtf(v2[o] + BN_EPSF);
    const float sh = b2[o] - m2[o] * sc;
    const float bo = bf2[o];
#pragma unroll
    for (int ti = 0; ti < 2; ++ti)
#pragma unroll
      for (int i = 0; i < 8; ++i) {
        int n = ti*16 + half*8 + i;
        float v = Hb[(size_t)n*CCH + o] + acc[ti][tj][i] + bo;
        ob[(size_t)n*CCH + o] = v * sc + sh;
      }
  }
}

// ---------------------------------------------------------------------------
extern "C" void kernel_launch(void* const* d_in, const int* in_sizes, int n_in,
                              void* d_out, int out_size, void* d_ws, size_t ws_size,
                              hipStream_t stream) {
  (void)in_sizes; (void)n_in; (void)out_size; (void)ws_size;
  const float* x  = (const float*)d_in[0];
  const float* Wq = (const float*)d_in[1];
  const float* Wk = (const float*)d_in[2];
  const float* Wv = (const float*)d_in[3];
  const float* bv = (const float*)d_in[4];
  const float* Wt = (const float*)d_in[5];
  const float* bt = (const float*)d_in[6];
  const float* g1 = (const float*)d_in[7];
  const float* b1 = (const float*)d_in[8];
  const float* m1 = (const float*)d_in[9];
  const float* v1 = (const float*)d_in[10];
  const float* W1 = (const float*)d_in[11];
  const float* bf1= (const float*)d_in[12];
  const float* W2 = (const float*)d_in[13];
  const float* bf2= (const float*)d_in[14];
  const float* g2 = (const float*)d_in[15];
  const float* b2 = (const float*)d_in[16];
  const float* m2 = (const float*)d_in[17];
  const float* v2 = (const float*)d_in[18];
  float* out = (float*)d_out;

  float* ws     = (float*)d_ws;
  float* Q      = ws;                                 // B*N*32
  float* Kf     = Q      + (size_t)BATCH*NPTS*DQK;    // B*N*32
  float* V      = Kf     + (size_t)BATCH*NPTS*DQK;    // B*N*128
  float* rowmax = V      + (size_t)BATCH*NPTS*CCH;    // B*N
  float* rowsum = rowmax + (size_t)BATCH*NPTS;        // B*N
  float* U      = rowsum + (size_t)BATCH*NPTS;        // B*N*128
  float* H      = U      + (size_t)BATCH*NPTS*CCH;    // B*N*128
  float* Ff     = H      + (size_t)BATCH*NPTS*CCH;    // B*N*512

  qkv_kernel  <<<dim3(NPTS/32, 6,       BATCH), 32,  0, stream>>>(x, Wq, Wk, Wv, bv, Q, Kf, V);
  stats_kernel<<<dim3(NPTS/16, BATCH),          256, 0, stream>>>(Q, Kf, rowmax, rowsum);
  attnv_kernel<<<dim3(NPTS/16, BATCH),          32,  0, stream>>>(Q, Kf, V, x, rowmax, rowsum, U);
  trans_kernel<<<dim3(NPTS/32, CCH/32,  BATCH), 32,  0, stream>>>(U, Wt, bt, g1, b1, m1, v1, x, H);
  ffn1_kernel <<<dim3(NPTS/32, FDIM/32, BATCH), 32,  0, stream>>>(H, W1, bf1, Ff);
  ffn2_kernel <<<dim3(NPTS/32, CCH/32,  BATCH), 32,  0, stream>>>(Ff, W2, bf2, H, g2, b2, m2, v2, out);
}